// GraphWaveletTransform_36051955482933
// MI455X (gfx1250) — compile-verified
//
#include <hip/hip_runtime.h>

typedef __attribute__((ext_vector_type(2))) float v2f;
typedef __attribute__((ext_vector_type(8))) float v8f;

// ===========================================================================
// Stage 1: build CSR-by-destination once per call (dst is static across all
// 4 diffusion steps). Kills the 205M-global-atomic hot path of COO scatter.
// ===========================================================================

__global__ void __launch_bounds__(256)
hist_kernel(const int* __restrict__ dst, int* __restrict__ cnt, int n_edges) {
    int e = blockIdx.x * blockDim.x + threadIdx.x;
    if (e < n_edges) atomicAdd(&cnt[dst[e]], 1);
}

// Exclusive scan of cnt[0..n) into rowptr[0..n], rowptr[n] = total.
// Single block of 1024 threads; LDS Hillis-Steele per 1024-chunk with carry.
__global__ void __launch_bounds__(1024)
scan_kernel(const int* __restrict__ cnt, int* __restrict__ rowptr, int n) {
    __shared__ int buf[1024];
    __shared__ int carry_s;
    const int tid = threadIdx.x;
    if (tid == 0) carry_s = 0;
    __syncthreads();

    for (int base = 0; base < n; base += 1024) {
        const int i = base + tid;
        const int v = (i < n) ? cnt[i] : 0;
        const int carry = carry_s;           // stable since prev iter's sync
        buf[tid] = v;
        __syncthreads();
#pragma unroll
        for (int off = 1; off < 1024; off <<= 1) {
            int t = (tid >= off) ? buf[tid - off] : 0;
            __syncthreads();
            buf[tid] += t;
            __syncthreads();
        }
        const int incl = buf[tid];
        if (i < n) rowptr[i] = carry + incl - v;   // exclusive
        __syncthreads();                            // all carry reads done
        if (tid == 1023) carry_s = carry + incl;
        __syncthreads();
    }
    if (tid == 0) rowptr[n] = carry_s;
}

__global__ void __launch_bounds__(256)
scatter_kernel(const int* __restrict__ src, const int* __restrict__ dst,
               const float* __restrict__ w, int* __restrict__ cursor,
               int* __restrict__ src_s, float* __restrict__ w_s, int n_edges) {
    int e = blockIdx.x * blockDim.x + threadIdx.x;
    if (e >= n_edges) return;
    int pos = atomicAdd(&cursor[dst[e]], 1);
    src_s[pos] = src[e];
    w_s[pos]   = w[e];
}

// ===========================================================================
// Stage 2: CSR SpMM  y[d] = sum_{e in row d} w_e * x[src_e]   ([N,64] f32)
// One wave per destination node; lane owns feature pair lane*2..+1.
// Gathers are coalesced 256B row reads; accumulation in registers; single
// store per row. Zero atomics, no output pre-zeroing needed.
// ===========================================================================
__global__ void __launch_bounds__(256)
spmm_csr_kernel(const float* __restrict__ x, float* __restrict__ y,
                const int* __restrict__ rowptr, const int* __restrict__ src_s,
                const float* __restrict__ w_s, int n) {
    const int node = (blockIdx.x * blockDim.x + threadIdx.x) >> 5;
    if (node >= n) return;
    const int lane = threadIdx.x & 31;

    const int beg = rowptr[node];
    const int end = rowptr[node + 1];

    float ax = 0.0f, ay = 0.0f;
    int i = beg;
    // 2-way unroll: two independent gathers in flight per iteration
    for (; i + 1 < end; i += 2) {
        int   s0 = src_s[i],     s1 = src_s[i + 1];
        float w0 = w_s[i],       w1 = w_s[i + 1];
        float2 v0 = *(const float2*)(x + (size_t)s0 * 64 + lane * 2);
        float2 v1 = *(const float2*)(x + (size_t)s1 * 64 + lane * 2);
        ax = fmaf(w0, v0.x, ax); ay = fmaf(w0, v0.y, ay);
        ax = fmaf(w1, v1.x, ax); ay = fmaf(w1, v1.y, ay);
    }
    if (i < end) {
        int   s0 = src_s[i];
        float w0 = w_s[i];
        float2 v0 = *(const float2*)(x + (size_t)s0 * 64 + lane * 2);
        ax = fmaf(w0, v0.x, ax); ay = fmaf(w0, v0.y, ay);
    }

    float2 o; o.x = ax; o.y = ay;
    *(float2*)(y + (size_t)node * 64 + lane * 2) = o;
}

// ===========================================================================
// Stage 3: fused feats-assembly + GEMM + bias + PReLU (V_WMMA_F32_16X16X4_F32)
// feats [N,384] blocks (k>>6): 0:x4 | 1:|x1-x2| | 2:|x2-x4| | 3:|x1-x2|
//                              4:|x3-x2| | 5:|x4-x3|   (blocks 1+3 fused)
// ===========================================================================
#define BS_STRIDE 65

template<bool DIFF, int NB>
__device__ __forceinline__ void gemm_pass(
    const float* __restrict__ P, const float* __restrict__ Q,
    const float* __restrict__ fc_w, int bcol0, int bcol1,
    float* __restrict__ Bs, int tid, int r, int lhalf, size_t rowoff,
    v8f* acc) {
    for (int i = tid; i < NB * 4096; i += 256) {
        int chunk = i >> 12;
        int rem   = i & 4095;
        int nn    = rem >> 6;
        int kk    = rem & 63;
        int bc    = (chunk == 0) ? bcol0 : bcol1;
        Bs[(chunk * 64 + kk) * BS_STRIDE + nn] = fc_w[(size_t)nn * 384 + bc + kk];
    }
    __syncthreads();

#pragma unroll 4
    for (int k0 = 0; k0 < 64; k0 += 4) {
        const int k = k0 + lhalf * 2;          // A layout: lanes 16-31 hold K+2
        float2 pv = *(const float2*)(P + rowoff + k);
        v2f a;
        if (DIFF) {
            float2 qv = *(const float2*)(Q + rowoff + k);
            a.x = fabsf(pv.x - qv.x);
            a.y = fabsf(pv.y - qv.y);
        } else {
            a.x = pv.x;
            a.y = pv.y;
        }
#pragma unroll
        for (int c = 0; c < NB; ++c) {
#pragma unroll
            for (int t = 0; t < 4; ++t) {
                v2f b;
                b.x = Bs[(c * 64 + k)     * BS_STRIDE + t * 16 + r];
                b.y = Bs[(c * 64 + k + 1) * BS_STRIDE + t * 16 + r];
                acc[t] = __builtin_amdgcn_wmma_f32_16x16x4_f32(
                    false, a, false, b, (short)0, acc[t], false, false);
            }
        }
    }
    __syncthreads();
}

__global__ void __launch_bounds__(256)
wgw_gemm_kernel(const float* __restrict__ x1, const float* __restrict__ x2,
                const float* __restrict__ x3, const float* __restrict__ x4,
                const float* __restrict__ fc_w, const float* __restrict__ bias,
                const float* __restrict__ prelu_a, float* __restrict__ out, int n) {
    __shared__ float Bs[128 * BS_STRIDE];      // widest pass: 2 K-slices (33 KB)

    const int tid   = threadIdx.x;
    const int lane  = tid & 31;
    const int wave  = tid >> 5;
    const int lhalf = lane >> 4;
    const int r     = lane & 15;

    const int    tile   = blockIdx.x * 8 + wave;
    const int    row0   = tile * 16;
    const int    rowc   = (row0 + r < n) ? (row0 + r) : (n - 1);
    const size_t rowoff = (size_t)rowc * 64;
    const float  slope  = *prelu_a;

    v8f acc[4] = {};

    gemm_pass<false, 1>(x4, x4, fc_w,   0,   0, Bs, tid, r, lhalf, rowoff, acc);
    gemm_pass<true,  2>(x1, x2, fc_w,  64, 192, Bs, tid, r, lhalf, rowoff, acc);
    gemm_pass<true,  1>(x2, x4, fc_w, 128,   0, Bs, tid, r, lhalf, rowoff, acc);
    gemm_pass<true,  1>(x3, x2, fc_w, 256,   0, Bs, tid, r, lhalf, rowoff, acc);
    gemm_pass<true,  1>(x4, x3, fc_w, 320,   0, Bs, tid, r, lhalf, rowoff, acc);

    if (row0 < n) {
#pragma unroll
        for (int t = 0; t < 4; ++t) {
#pragma unroll
            for (int v = 0; v < 8; ++v) {
                int m  = v + lhalf * 8;
                int rg = row0 + m;
                int cg = t * 16 + r;
                if (rg < n) {
                    float val = acc[t][v] + bias[cg];
                    out[(size_t)rg * 64 + cg] = (val > 0.0f) ? val : slope * val;
                }
            }
        }
    }
}

// ===========================================================================
// Launch: counting-sort by dst (once) -> 4 CSR SpMMs -> fused WMMA GEMM.
// Workspace: 4*N*64 f32 (51.2MB) + CSR arrays (~13MB).
// ===========================================================================
extern "C" void kernel_launch(void* const* d_in, const int* in_sizes, int n_in,
                              void* d_out, int out_size, void* d_ws, size_t ws_size,
                              hipStream_t stream) {
    const float* seq     = (const float*)d_in[0];   // [N,64]
    const int*   eidx    = (const int*)  d_in[1];   // [2,E]
    const float* ew      = (const float*)d_in[2];   // [E]
    const float* fc_w    = (const float*)d_in[3];   // [64,384]
    const float* bias    = (const float*)d_in[4];   // [64]
    const float* prelu_a = (const float*)d_in[5];   // [1]

    const int N = in_sizes[0] / 64;
    const int E = in_sizes[2];
    const int* src = eidx;
    const int* dst = eidx + E;

    // Workspace layout (16B-aligned slabs)
    char* p = (char*)d_ws;
    auto alloc = [&](size_t bytes) {
        char* q = p;
        p += (bytes + 15) & ~(size_t)15;
        return q;
    };
    float* x1      = (float*)alloc((size_t)N * 64 * sizeof(float));
    float* x2      = (float*)alloc((size_t)N * 64 * sizeof(float));
    float* x3      = (float*)alloc((size_t)N * 64 * sizeof(float));
    float* x4      = (float*)alloc((size_t)N * 64 * sizeof(float));
    int*   cnt     = (int*)  alloc((size_t)N * sizeof(int));
    int*   rowptr  = (int*)  alloc((size_t)(N + 1) * sizeof(int));
    int*   cursor  = (int*)  alloc((size_t)N * sizeof(int));
    int*   src_s   = (int*)  alloc((size_t)E * sizeof(int));
    float* w_s     = (float*)alloc((size_t)E * sizeof(float));

    // --- build CSR by destination (reused by all 4 SpMMs) ---
    hipMemsetAsync(cnt, 0, (size_t)N * sizeof(int), stream);
    const int eblocks = (E + 255) / 256;
    hist_kernel<<<eblocks, 256, 0, stream>>>(dst, cnt, E);
    scan_kernel<<<1, 1024, 0, stream>>>(cnt, rowptr, N);
    hipMemcpyAsync(cursor, rowptr, (size_t)N * sizeof(int),
                   hipMemcpyDeviceToDevice, stream);
    scatter_kernel<<<eblocks, 256, 0, stream>>>(src, dst, ew, cursor, src_s, w_s, E);

    // --- 4 diffusion steps, atomic-free ---
    const int sblocks = (N * 32 + 255) / 256;
    spmm_csr_kernel<<<sblocks, 256, 0, stream>>>(seq, x1, rowptr, src_s, w_s, N);
    spmm_csr_kernel<<<sblocks, 256, 0, stream>>>(x1,  x2, rowptr, src_s, w_s, N);
    spmm_csr_kernel<<<sblocks, 256, 0, stream>>>(x2,  x3, rowptr, src_s, w_s, N);
    spmm_csr_kernel<<<sblocks, 256, 0, stream>>>(x3,  x4, rowptr, src_s, w_s, N);

    // --- fused feats + GEMM + bias + PReLU ---
    const int tiles  = (N + 15) / 16;
    const int blocks = (tiles + 7) / 8;
    wgw_gemm_kernel<<<blocks, 256, 0, stream>>>(x1, x2, x3, x4, fc_w, bias,
                                                prelu_a, (float*)d_out, N);
}